// QuantizationLayer_25795573580345
// MI455X (gfx1250) — compile-verified
//
#include <hip/hip_runtime.h>

typedef __attribute__((ext_vector_type(2))) float v2f;
typedef __attribute__((ext_vector_type(8))) float v8f;

#define C_BINS 9
#define HID    100
#define HPAD   112          // 7 x 16 column tiles (zero padded)
#define NTILES 7
#define KPAIRS 50           // HID/2, K handled in pairs per WMMA half
#define W_IMG  336
#define H_IMG  256
#define WH     (W_IMG * H_IMG)        // 86016
#define WHC    (WH * C_BINS)          // 774144
#define WHC2   (2 * WHC)              // 1548288

__global__ void zero_out_kernel(float* __restrict__ out, int n) {
  int i = blockIdx.x * blockDim.x + threadIdx.x;
  int stride = gridDim.x * blockDim.x;
  for (; i < n; i += stride) out[i] = 0.0f;
}

__global__ __launch_bounds__(256, 4) void quant_voxel_kernel(
    const float* __restrict__ events, int Nev,
    const float* __restrict__ w1, const float* __restrict__ b1,
    const float* __restrict__ w2, const float* __restrict__ b2,
    const float* __restrict__ w3, const float* __restrict__ b3,
    float* __restrict__ out)
{
  // LDS staging of weights (shared by all 8 waves of the block)
  __shared__ float2 W2p[KPAIRS * HPAD];  // {W2[2kp][n], W2[2kp+1][n]}  44.8 KB
  __shared__ float4 wb1p[KPAIRS];        // {w1[2kp], b1[2kp], w1[2kp+1], b1[2kp+1]}
  __shared__ float  b2s[HPAD];
  __shared__ float  w3s[HPAD];
  __shared__ float  b3s;

  const int tid = threadIdx.x;
  for (int i = tid; i < KPAIRS * HPAD; i += 256) {
    int kp = i / HPAD, col = i - kp * HPAD;
    float2 v;
    if (col < HID) {
      v.x = w2[(2 * kp)     * HID + col];
      v.y = w2[(2 * kp + 1) * HID + col];
    } else { v.x = 0.0f; v.y = 0.0f; }
    W2p[i] = v;
  }
  if (tid < KPAIRS) {
    float4 v;
    v.x = w1[2 * tid];     v.y = b1[2 * tid];
    v.z = w1[2 * tid + 1]; v.w = b1[2 * tid + 1];
    wb1p[tid] = v;
  }
  if (tid < HPAD) {
    b2s[tid] = (tid < HID) ? b2[tid] : 0.0f;
    w3s[tid] = (tid < HID) ? w3[tid] : 0.0f;
  }
  if (tid == 0) b3s = b3[0];
  __syncthreads();   // single block-wide barrier; per-wave loop below is barrier-free

  const int lane = tid & 31;
  const int lm   = lane & 15;          // column-within-tile / row id
  const int kh   = lane >> 4;          // which half of the wave (K sub-pair)

  // hoisted per-lane epilogue weights (invariant across tiles)
  float b2v[NTILES], w3v[NTILES];
#pragma unroll
  for (int nt = 0; nt < NTILES; ++nt) {
    b2v[nt] = b2s[nt * 16 + lm];
    w3v[nt] = w3s[nt * 16 + lm];
  }
  const float b3v = b3s;

  const long long total = (long long)Nev * C_BINS;     // 18M evals
  const int numTiles = (int)((total + 15) / 16);
  const int wave   = blockIdx.x * (blockDim.x >> 5) + (tid >> 5);
  const int nwaves = gridDim.x * (blockDim.x >> 5);

  for (int tile = wave; tile < numTiles; tile += nwaves) {
    const int base = tile * 16;        // eval-id of row 0 (bin-major: g = bin*Nev + n)
    // per-lane row setup (rows duplicated across the two wave halves)
    int g = base + lm;
    if (g >= (int)total) g = (int)total - 1;
    int binL = g / Nev;
    int nrow = g - binL * Nev;
    const float tval = events[(size_t)nrow * 5 + 2];
    const float s = tval - (float)binL * 0.125f;       // MLP scalar input

    v8f c[NTILES];
    const v8f zero8 = {0.f, 0.f, 0.f, 0.f, 0.f, 0.f, 0.f, 0.f};
#pragma unroll
    for (int nt = 0; nt < NTILES; ++nt) c[nt] = zero8;

    // K loop: 25 steps of K=4; kp = K-pair owned by this half of the wave.
    // unroll-by-5: enough independent DS loads to hide LDS latency without
    // ballooning register pressure / spilling (keep operands in LDS).
#pragma unroll 5
    for (int k = 0; k < 25; ++k) {
      const int kp = 2 * k + kh;       // 0..49
      // A fragment: layer-1 activations computed in registers
      const float4 wb = wb1p[kp];      // ds_load_b128 broadcast within each half
      float ax = fmaf(s, wb.x, wb.y); ax = ax < 0.f ? 0.1f * ax : ax;
      float ay = fmaf(s, wb.z, wb.w); ay = ay < 0.f ? 0.1f * ay : ay;
      v2f a; a.x = ax; a.y = ay;
      const float2* Wrow = &W2p[kp * HPAD + lm];
#pragma unroll
      for (int nt = 0; nt < NTILES; ++nt) {
        const float2 wv = Wrow[nt * 16];   // one ds_load_b64, conflict-free
        v2f bf; bf.x = wv.x; bf.y = wv.y;
        c[nt] = __builtin_amdgcn_wmma_f32_16x16x4_f32(
            false, a, false, bf, (short)0, c[nt], false, false);
      }
    }

    // Epilogue: h2 = leaky(c + b2); per-lane partial dot with w3
    float partial[8];
#pragma unroll
    for (int r = 0; r < 8; ++r) partial[r] = 0.0f;
#pragma unroll
    for (int nt = 0; nt < NTILES; ++nt) {
#pragma unroll
      for (int r = 0; r < 8; ++r) {
        float h = c[nt][r] + b2v[nt];
        h = h < 0.f ? 0.1f * h : h;
        partial[r] = fmaf(h, w3v[nt], partial[r]);
      }
    }
    // reduce across the 16 lanes of each half (rows 0-7 low half, 8-15 high half)
#pragma unroll
    for (int r = 0; r < 8; ++r) {
      float v = partial[r];
      v += __shfl_xor(v, 1, 32);
      v += __shfl_xor(v, 2, 32);
      v += __shfl_xor(v, 4, 32);
      v += __shfl_xor(v, 8, 32);
      partial[r] = v;
    }
    // lane (lane&7)==r of each half picks row sum r (static select tree)
    float S = partial[0];
#pragma unroll
    for (int r = 1; r < 8; ++r) S = ((lane & 7) == r) ? partial[r] : S;

    if ((lane & 15) < 8) {                       // 16 active scatter lanes
      const int M = (lane & 7) + (kh << 3);      // row 0..15
      const long long gM = (long long)base + M;
      if (gM < total) {
        const int binM = (int)(gM / Nev);
        const int nM   = (int)(gM - (long long)binM * Nev);
        const float* ev = events + (size_t)nM * 5;
        const float xf = ev[0], yf = ev[1], tm = ev[2], pf = ev[3], bf = ev[4];
        const int p01 = pf > 0.f ? 1 : 0;
        const int idx = (int)xf + W_IMG * (int)yf + WHC * p01
                      + WHC2 * (int)bf + WH * binM;
        const float val = tm * (S + b3v);
        __hip_atomic_fetch_add(&out[idx], val, __ATOMIC_RELAXED,
                               __HIP_MEMORY_SCOPE_AGENT);
      }
    }
  }
}

extern "C" void kernel_launch(void* const* d_in, const int* in_sizes, int n_in,
                              void* d_out, int out_size, void* d_ws, size_t ws_size,
                              hipStream_t stream) {
  (void)n_in; (void)d_ws; (void)ws_size;
  const float* events = (const float*)d_in[0];
  // d_in[1..3] = H, W, B (fixed 256/336/8 per reference, hardcoded)
  const float* w1 = (const float*)d_in[4];
  const float* b1 = (const float*)d_in[5];
  const float* w2 = (const float*)d_in[6];
  const float* b2 = (const float*)d_in[7];
  const float* w3 = (const float*)d_in[8];
  const float* b3 = (const float*)d_in[9];
  float* out = (float*)d_out;

  const int Nev = in_sizes[0] / 5;

  zero_out_kernel<<<4096, 256, 0, stream>>>(out, out_size);
  quant_voxel_kernel<<<2048, 256, 0, stream>>>(events, Nev,
                                               w1, b1, w2, b2, w3, b3, out);
}